// GlobalGNN_21663815041270
// MI455X (gfx1250) — compile-verified
//
#include <hip/hip_runtime.h>
#include <hip/hip_bf16.h>

#define HID 128
#define TILES_PER_BLOCK 8                       // 8 x 16 = 128 rows per block

typedef __attribute__((ext_vector_type(2))) float v2f;
typedef __attribute__((ext_vector_type(8))) float v8f;

// ---------------------------------------------------------------------------
// Zero the aggregation buffer (float4 stores, graph-capture safe).
// ---------------------------------------------------------------------------
__global__ __launch_bounds__(256) void gnn_zero_kernel(float4* __restrict__ p, long n4) {
    long i = (long)blockIdx.x * blockDim.x + threadIdx.x;
    if (i < n4) p[i] = make_float4(0.f, 0.f, 0.f, 0.f);
}

// ---------------------------------------------------------------------------
// SpMM scatter: one wave32 per edge. Each lane owns a float4 slice of the
// 128-wide feature row (32 lanes * 4 = 128). h row gathers hit L2 (h = 51MB
// << 192MB L2); scatter uses hardware global_atomic_add_f32.
// ---------------------------------------------------------------------------
__global__ __launch_bounds__(256) void gnn_spmm_kernel(const int* __restrict__ erow,
                                                       const int* __restrict__ ecol,
                                                       const float* __restrict__ ew,
                                                       const float* __restrict__ h,
                                                       float* __restrict__ agg,
                                                       int E) {
    const int e    = (int)blockIdx.x * 8 + (int)(threadIdx.x >> 5);
    const int lane = threadIdx.x & 31;
    if (e >= E) return;

    const int   c  = ecol[e];
    const int   r  = erow[e];
    const float wt = ew[e];

    const float4 v = ((const float4*)(h + (size_t)c * HID))[lane];
    float* dst = agg + (size_t)r * HID + lane * 4;

    unsafeAtomicAdd(dst + 0, v.x * wt);
    unsafeAtomicAdd(dst + 1, v.y * wt);
    unsafeAtomicAdd(dst + 2, v.z * wt);
    unsafeAtomicAdd(dst + 3, v.w * wt);
}

// ---------------------------------------------------------------------------
// Dense epilogue: out = relu(agg @ W^T + b) via V_WMMA_F32_16X16X4_F32.
// Each block owns 128 rows (8 x 16-row tiles): W (64KB) is staged in LDS
// ONCE per 128 rows (8x less L2 traffic than per-tile staging), all 8
// A-tiles (64KB) staged together, single barrier, then 8 x 32 chained WMMAs
// per wave. 8 waves/block; wave w owns output columns [16w, 16w+16).
// ---------------------------------------------------------------------------
__global__ __launch_bounds__(256) void gnn_gemm_relu_kernel(const float* __restrict__ agg,
                                                            const float* __restrict__ W,
                                                            const float* __restrict__ bias,
                                                            float* __restrict__ out,
                                                            int N) {
    __shared__ float Ws[HID * HID];                   // 64 KB : full weight matrix
    __shared__ float As[TILES_PER_BLOCK * 16 * HID];  // 64 KB : 128 rows of agg

    const int  tid = threadIdx.x;
    const long blk = blockIdx.x;

    // Stage W: 16384 floats = 4096 float4, 16 per thread.
    {
        const float4* src = (const float4*)W;
        float4*       dst = (float4*)Ws;
#pragma unroll
        for (int i = 0; i < 16; ++i) dst[tid + 256 * i] = src[tid + 256 * i];
    }
    // Stage 128 rows of A: 16384 floats = 4096 float4, 16 per thread.
    // Clamp (not branch) out-of-range indices so the ragged last block keeps
    // every lane active ahead of the WMMAs (EXEC must be all ones).
    {
        const float4* src  = (const float4*)agg;
        float4*       dst  = (float4*)As;
        const long    maxi = (long)N * (HID / 4) - 1;
#pragma unroll
        for (int i = 0; i < 16; ++i) {
            long gi = blk * 4096 + i * 256 + tid;
            dst[tid + 256 * i] = src[gi <= maxi ? gi : maxi];
        }
    }
    __syncthreads();

    const int lane = tid & 31;
    const int wv   = tid >> 5;              // wave id -> N-tile 0..7
    const int m    = lane & 15;             // row within A tile / col within N tile
    const int koff = (lane >> 4) << 1;      // lanes 0-15: K+{0,1}; lanes 16-31: K+{2,3}
    const int ncol = (wv << 4) + m;         // global output column 0..127

    const float bv = bias[ncol];
    const float* bp0 = &Ws[(ncol << 7) + koff];

#pragma unroll 1
    for (int t = 0; t < TILES_PER_BLOCK; ++t) {
        // A: 16x4 f32 per WMMA -> lane holds A[m][k0+koff], A[m][k0+koff+1]
        // B: 4x16 f32, B[k][n] = W[n][k] -> lane holds W[ncol][k0+koff], +1
        const float* ap = &As[t * 16 * HID + (m << 7) + koff];
        const float* bp = bp0;

        v8f c = {};
#pragma unroll
        for (int k0 = 0; k0 < HID; k0 += 4) {
            v2f a = *(const v2f*)ap; ap += 4;   // ds_load_b64 (8B aligned)
            v2f b = *(const v2f*)bp; bp += 4;
            c = __builtin_amdgcn_wmma_f32_16x16x4_f32(
                    /*neg_a=*/false, a, /*neg_b=*/false, b,
                    /*c_mod=*/(short)0, c, /*reuse_a=*/false, /*reuse_b=*/false);
        }

        // C/D layout: VGPR v -> M = v (lanes 0-15) or v+8 (lanes 16-31),
        // N = lane&15.
        const long rowbase = blk * (TILES_PER_BLOCK * 16) + t * 16 + ((lane >> 4) << 3);
#pragma unroll
        for (int v = 0; v < 8; ++v) {
            const long row = rowbase + v;
            if (row < N) {
                float r = c[v] + bv;
                out[row * HID + ncol] = r > 0.f ? r : 0.f;
            }
        }
    }
}

// ---------------------------------------------------------------------------
// Orchestration: agg lives in d_ws; the step-1 hidden state reuses d_out
// (same shape as final output), so only one 51.2MB scratch buffer is needed.
// ---------------------------------------------------------------------------
extern "C" void kernel_launch(void* const* d_in, const int* in_sizes, int n_in,
                              void* d_out, int out_size, void* d_ws, size_t ws_size,
                              hipStream_t stream) {
    const int*   erow = (const int*)d_in[0];
    const int*   ecol = (const int*)d_in[1];
    const float* ew   = (const float*)d_in[2];
    const float* h0   = (const float*)d_in[3];
    const float* W    = (const float*)d_in[4];
    const float* bias = (const float*)d_in[5];

    const int E = in_sizes[0];
    const int N = in_sizes[3] / HID;

    float* agg = (float*)d_ws;
    float* out = (float*)d_out;

    const long n4       = (long)N * HID / 4;
    const int  zeroGrid = (int)((n4 + 255) / 256);
    const int  spmmGrid = (E + 7) / 8;                              // wave32 per edge
    const int  gemmGrid = (N + TILES_PER_BLOCK * 16 - 1) / (TILES_PER_BLOCK * 16);

    // Step 1: h0 -> agg -> out
    gnn_zero_kernel<<<zeroGrid, 256, 0, stream>>>((float4*)agg, n4);
    gnn_spmm_kernel<<<spmmGrid, 256, 0, stream>>>(erow, ecol, ew, h0, agg, E);
    gnn_gemm_relu_kernel<<<gemmGrid, 256, 0, stream>>>(agg, W, bias, out, N);

    // Step 2: out -> agg -> out
    gnn_zero_kernel<<<zeroGrid, 256, 0, stream>>>((float4*)agg, n4);
    gnn_spmm_kernel<<<spmmGrid, 256, 0, stream>>>(erow, ecol, ew, out, agg, E);
    gnn_gemm_relu_kernel<<<gemmGrid, 256, 0, stream>>>(agg, W, bias, out, N);
}